// SelfAttention_463856468090
// MI455X (gfx1250) — compile-verified
//
#include <hip/hip_runtime.h>
#include <stdint.h>

typedef __attribute__((ext_vector_type(2))) float v2f;
typedef __attribute__((ext_vector_type(8))) float v8f;

#define B_DIM 4
#define S_DIM 2048
#define H_DIM 16
#define D_DIM 64
#define NEG_VAL (-10000.0f)
#define SROW (3 * H_DIM * D_DIM)   /* 3072 floats between consecutive s in qkv */
#define TILES (S_DIM / 16)         /* 128 key tiles */
#define SCALE 0.125f               /* 1/sqrt(64) */

// Swap data between lane L and lane L^16 (crosses DPP16 rows): ds_swizzle
// group-of-32 mode, offset = (xor=0x10)<<10 | (or=0)<<5 | (and=0x1f) = 0x401F.
__device__ __forceinline__ float ds_swapx16_f32(float x) {
    int r = __builtin_amdgcn_ds_swizzle(__builtin_bit_cast(int, x), 0x401F);
    return __builtin_bit_cast(float, r);
}

// Gather: dst[lane] = src[idx_bytes/4]
__device__ __forceinline__ float ds_bperm_f32(int idx_bytes, float x) {
    int r = __builtin_amdgcn_ds_bpermute(idx_bytes, __builtin_bit_cast(int, x));
    return __builtin_bit_cast(float, r);
}

// Flash attention: one wave32 owns a 16-query tile of one (b,h).
// QK^T and PV run on V_WMMA_F32_16X16X4_F32 (fp32-faithful to the reference).
// Softmax runs in the transposed (A-layout) domain: one row per lane.
__global__ __launch_bounds__(256) void fa_wmma_f32_kernel(
    const float* __restrict__ qkv,
    const unsigned char* __restrict__ key_padding_mask,
    float* __restrict__ out)
{
    __shared__ float lds[8][16 * 18];  // per-wave 16x16 score tile, stride 18 (8B aligned, conflict-free)

    const int lane = threadIdx.x & 31;
    const int wib  = threadIdx.x >> 5;          // wave in block (0..7)
    const int wid  = blockIdx.x * 8 + wib;      // 0..8191
    const int tile = wid % TILES;               // query tile index
    const int bh   = wid / TILES;
    const int h    = bh % H_DIM;
    const int b    = bh / H_DIM;

    const int lr = lane & 15;                   // row (A side) / col (B,C side)
    const int hi = lane >> 4;                   // half-wave select

    const size_t baseQ = (size_t)b * S_DIM * SROW + 0 * (H_DIM * D_DIM) + (size_t)h * D_DIM;
    const size_t baseK = (size_t)b * S_DIM * SROW + 1 * (H_DIM * D_DIM) + (size_t)h * D_DIM;
    const size_t baseV = (size_t)b * S_DIM * SROW + 2 * (H_DIM * D_DIM) + (size_t)h * D_DIM;

    float* myLds = lds[wib];

    // bpermute byte-indices: C-layout vgpr j needs the A-domain value of row j+8*hi
    int bidx[8];
#pragma unroll
    for (int j = 0; j < 8; ++j) bidx[j] = (j + 8 * hi) * 4;

    // ---- Preload Q tile (16x64) in WMMA A-layout, pre-scaled by 1/sqrt(D)
    const int qrow = tile * 16 + lr;
    v2f qa[16];
    {
        const float* qp = qkv + baseQ + (size_t)qrow * SROW + 2 * hi;
#pragma unroll
        for (int kk = 0; kk < 16; ++kk) {
            v2f q = *(const v2f*)(qp + 4 * kk);
            q.x *= SCALE;
            q.y *= SCALE;
            qa[kk] = q;
        }
    }

    // ---- Online-softmax state (A-domain: this lane owns row lane&15)
    float mrow_a = -3.0e38f;
    float lrow_a = 0.0f;

    // ---- Output accumulators (16x64 as 4 N-tiles of 16x16, C-layout)
    v8f acc[4];
#pragma unroll
    for (int t = 0; t < 4; ++t)
#pragma unroll
        for (int j = 0; j < 8; ++j) acc[t][j] = 0.0f;

    for (int kt = 0; kt < TILES; ++kt) {
        const int kb = kt * 16;

        // ---- K tile (16x64) in B-layout: same float2 pattern as Q (N = lr)
        v2f kbuf[16];
        {
            const float* kp = qkv + baseK + (size_t)(kb + lr) * SROW + 2 * hi;
#pragma unroll
            for (int kk = 0; kk < 16; ++kk)
                kbuf[kk] = *(const v2f*)(kp + 4 * kk);
        }

        // ---- V tile in B-layout for PV: vb[kk][t].{x,y} = V[kb+4kk+2hi+{0,1}][16t+lr]
        v2f vb[4][4];
#pragma unroll
        for (int kk = 0; kk < 4; ++kk) {
            const float* vp = qkv + baseV + (size_t)(kb + 4 * kk + 2 * hi) * SROW + lr;
#pragma unroll
            for (int t = 0; t < 4; ++t) {
                v2f v;
                v.x = vp[16 * t];
                v.y = vp[16 * t + SROW];
                vb[kk][t] = v;
            }
        }

        // ---- Scores: S = (Q*scale) * K^T (16 WMMAs over K=64), C-layout result
        v8f c;
#pragma unroll
        for (int j = 0; j < 8; ++j) c[j] = 0.0f;
#pragma unroll
        for (int kk = 0; kk < 16; ++kk)
            c = __builtin_amdgcn_wmma_f32_16x16x4_f32(
                    false, qa[kk], false, kbuf[kk], (short)0, c, false, false);

        // ---- key padding mask (column = kb + lr)
        const float pad = key_padding_mask[(size_t)b * S_DIM + kb + lr] ? 0.0f : NEG_VAL;
#pragma unroll
        for (int j = 0; j < 8; ++j) c[j] += pad;

        // ---- Transpose raw scores (C-layout -> A-layout) via per-wave LDS tile.
        // DS ops are in-order within a wave; compiler barrier + wait for safety.
#pragma unroll
        for (int j = 0; j < 8; ++j)
            myLds[(j + 8 * hi) * 18 + lr] = c[j];
        asm volatile("s_wait_dscnt 0" ::: "memory");
        v2f ct[4];
#pragma unroll
        for (int kk = 0; kk < 4; ++kk)
            ct[kk] = *(const v2f*)&myLds[lr * 18 + 4 * kk + 2 * hi];

        // ---- Row max in A-domain: 8 local cols here, other 8 in lane^16
        float rmax = fmaxf(fmaxf(fmaxf(ct[0].x, ct[0].y), fmaxf(ct[1].x, ct[1].y)),
                           fmaxf(fmaxf(ct[2].x, ct[2].y), fmaxf(ct[3].x, ct[3].y)));
        rmax = fmaxf(rmax, ds_swapx16_f32(rmax));

        const float mnew    = fmaxf(mrow_a, rmax);
        const float alpha_a = __expf(mrow_a - mnew);
        mrow_a = mnew;

        // ---- P = exp(S - mnew): computed directly in WMMA A-layout registers
        v2f pa[4];
#pragma unroll
        for (int kk = 0; kk < 4; ++kk) {
            pa[kk].x = __expf(ct[kk].x - mnew);
            pa[kk].y = __expf(ct[kk].y - mnew);
        }

        // ---- Row sum in A-domain
        float rsum = (pa[0].x + pa[0].y) + (pa[1].x + pa[1].y)
                   + (pa[2].x + pa[2].y) + (pa[3].x + pa[3].y);
        rsum += ds_swapx16_f32(rsum);
        lrow_a = lrow_a * alpha_a + rsum;

        // ---- Broadcast alpha to C-layout rows; rescale output accumulators
#pragma unroll
        for (int j = 0; j < 8; ++j) {
            const float ac = ds_bperm_f32(bidx[j], alpha_a);
#pragma unroll
            for (int t = 0; t < 4; ++t) acc[t][j] *= ac;
        }

        // ---- O += P * V (4 N-tiles x 4 K-steps)
#pragma unroll
        for (int t = 0; t < 4; ++t)
#pragma unroll
            for (int kk = 0; kk < 4; ++kk)
                acc[t] = __builtin_amdgcn_wmma_f32_16x16x4_f32(
                             false, pa[kk], false, vb[kk][t], (short)0, acc[t], false, false);
    }

    // ---- Epilogue: broadcast 1/lrow to C-layout, store out[b, s, h, d]
    const float inv_a = 1.0f / lrow_a;
#pragma unroll
    for (int j = 0; j < 8; ++j) {
        const float inv  = ds_bperm_f32(bidx[j], inv_a);
        const int   srow = tile * 16 + j + 8 * hi;
        float* op = out + (((size_t)b * S_DIM + srow) * H_DIM + h) * D_DIM + lr;
#pragma unroll
        for (int t = 0; t < 4; ++t)
            op[16 * t] = acc[t][j] * inv;
    }
}

extern "C" void kernel_launch(void* const* d_in, const int* in_sizes, int n_in,
                              void* d_out, int out_size, void* d_ws, size_t ws_size,
                              hipStream_t stream) {
    (void)in_sizes; (void)n_in; (void)out_size; (void)d_ws; (void)ws_size;
    const float*         qkv  = (const float*)d_in[0];
    const unsigned char* mask = (const unsigned char*)d_in[1];
    float*               out  = (float*)d_out;

    const int total_waves = B_DIM * H_DIM * TILES;      // 8192
    dim3 grid(total_waves / 8), block(256);             // 8 waves per block
    hipLaunchKernelGGL(fa_wmma_f32_kernel, grid, block, 0, stream, qkv, mask, out);
}